// Aggregationlayer_15135464751166
// MI455X (gfx1250) — compile-verified
//
#include <hip/hip_runtime.h>
#include <hip/hip_bf16.h>

typedef float v2f __attribute__((ext_vector_type(2)));
typedef float v8f __attribute__((ext_vector_type(8)));
typedef float f4  __attribute__((ext_vector_type(4)));

#define KNB   16
#define HID   256
#define AGG_K 512
#define CLIPV 1000.0f

#define ROWS_PER_BLK 32
#define AGG_STRIDE   516   // 512 + 4 pad: stride%64 == 4 -> conflict-free b64 row-strided reads

__device__ __forceinline__ float clipf(float v) {
    return fminf(fmaxf(v, -CLIPV), CLIPV);
}

__device__ __forceinline__ v8f wmma_f32(v2f a, v2f b, v8f c) {
    // D = A(16x4,f32) * B(4x16,f32) + C(16x16,f32)
    return __builtin_amdgcn_wmma_f32_16x16x4_f32(
        /*neg_a=*/false, a, /*neg_b=*/false, b,
        /*c_mod=*/(short)0, c, /*reuse_a=*/false, /*reuse_b=*/false);
}

__device__ __forceinline__ float silu_fast(float v) {
    // v * sigmoid(v) via v_exp_f32 + v_rcp_f32 (avoid the IEEE divide sequence)
    return v * __builtin_amdgcn_rcpf(1.0f + __expf(-v));
}

// -------- coord = clip(x) + mean_k clip(trans) --------
__global__ __launch_bounds__(256) void coord_kernel(
    const float* __restrict__ x, const float* __restrict__ trans,
    float* __restrict__ coord, int n)
{
    int i = blockIdx.x * blockDim.x + threadIdx.x;
    if (i >= n) return;
    const float* t = trans + (size_t)i * (KNB * 3);
    float s0 = 0.f, s1 = 0.f, s2 = 0.f;
#pragma unroll
    for (int k = 0; k < KNB; ++k) {
        s0 += clipf(t[k * 3 + 0]);
        s1 += clipf(t[k * 3 + 1]);
        s2 += clipf(t[k * 3 + 2]);
    }
    __builtin_nontemporal_store(clipf(x[i * 3 + 0]) + s0 * (1.0f / 16.0f), coord + i * 3 + 0);
    __builtin_nontemporal_store(clipf(x[i * 3 + 1]) + s1 * (1.0f / 16.0f), coord + i * 3 + 1);
    __builtin_nontemporal_store(clipf(x[i * 3 + 2]) + s2 * (1.0f / 16.0f), coord + i * 3 + 2);
}

// -------- h = hh + SiLU([hh | sum_k ef] @ W1 + b1) @ W2 + b2 --------
// Block: 256 threads = 8 waves; 32 node rows (2 WMMA M-tiles).
// smem[row][0:256]   = hh (kept for residual)
// smem[row][256:512] = sum_k edge_feature  (GEMM1 input)
// smem[row][256:516] = overlaid with SiLU mid-activations after GEMM1 barrier
__global__ __launch_bounds__(256) void mlp_kernel(
    const float* __restrict__ ef, const float* __restrict__ hh,
    const float* __restrict__ W1, const float* __restrict__ b1,
    const float* __restrict__ W2, const float* __restrict__ b2,
    float* __restrict__ out_h, int n)
{
    __shared__ float smem[ROWS_PER_BLK][AGG_STRIDE];

    const int tid  = threadIdx.x;
    const int lane = tid & 31;
    const int wave = tid >> 5;
    const int r0   = blockIdx.x * ROWS_PER_BLK;

    // ---- Stage 1: hh -> smem[:,0:256] ; sum_k ef -> smem[:,256:512] ----
    for (int t = tid; t < ROWS_PER_BLK * 64; t += 256) {
        int row = t >> 6;       // 0..31
        int c4  = t & 63;       // float4 group within 256 cols
        int g   = r0 + row;
        f4 v = (f4)0.0f;
        if (g < n) v = *(const f4*)(hh + (size_t)g * HID + c4 * 4);
        *(f4*)&smem[row][c4 * 4] = v;

        f4 s = (f4)0.0f;
        if (g < n) {
            const f4* p = (const f4*)(ef + (size_t)g * (KNB * HID)) + c4;
#pragma unroll
            for (int k = 0; k < KNB; ++k)
                s += __builtin_nontemporal_load(p + k * (HID / 4));  // stream once, keep out of L2
        }
        *(f4*)&smem[row][HID + c4 * 4] = s;
    }
    __syncthreads();

    // WMMA fragment coordinates (f32 16x16x4 layout)
    const int arow = lane & 15;            // A-frag row within tile
    const int kb   = (lane >> 4) * 2;      // A/B frag K sub-offset (0 or 2)
    const int colA = wave * 32 + arow;     // this wave's first N-column tile
    const int colB = colA + 16;            // second N-column tile

    // ---- GEMM1: [32x512] @ W1[512x256], K in steps of 4 ----
    v8f acc[2][2] = {};
    const float b1a = b1[colA], b1b = b1[colB];

#pragma unroll 4
    for (int k = 0; k < AGG_K; k += 4) {
        v2f a0 = *(const v2f*)&smem[arow][k + kb];        // ds_load_b64, conflict-free
        v2f a1 = *(const v2f*)&smem[16 + arow][k + kb];
        v2f w0, w1;
        w0.x = W1[(size_t)(k + kb) * HID + colA];
        w0.y = W1[(size_t)(k + kb + 1) * HID + colA];
        w1.x = W1[(size_t)(k + kb) * HID + colB];
        w1.y = W1[(size_t)(k + kb + 1) * HID + colB];
        acc[0][0] = wmma_f32(a0, w0, acc[0][0]);
        acc[1][0] = wmma_f32(a1, w0, acc[1][0]);
        acc[0][1] = wmma_f32(a0, w1, acc[0][1]);
        acc[1][1] = wmma_f32(a1, w1, acc[1][1]);
    }

    __syncthreads();   // everyone done reading smem[:,256:512] before overlaying mid

    // ---- Epilogue 1: bias + SiLU -> mid (overlaid at smem[:,256:516]) ----
    const int hhalf = lane >> 4;
#pragma unroll
    for (int rt = 0; rt < 2; ++rt) {
#pragma unroll
        for (int ct = 0; ct < 2; ++ct) {
            const int   col = (ct == 0) ? colA : colB;
            const float bb  = (ct == 0) ? b1a : b1b;
#pragma unroll
            for (int r = 0; r < 8; ++r) {
                smem[rt * 16 + r + 8 * hhalf][HID + col] = silu_fast(acc[rt][ct][r] + bb);
            }
        }
    }
    __syncthreads();

    // ---- GEMM2: mid[32x256] @ W2[256x256] ----
    v8f acc2[2][2] = {};
    const float b2a = b2[colA], b2b = b2[colB];

#pragma unroll 4
    for (int k = 0; k < HID; k += 4) {
        v2f a0 = *(const v2f*)&smem[arow][HID + k + kb];
        v2f a1 = *(const v2f*)&smem[16 + arow][HID + k + kb];
        v2f w0, w1;
        w0.x = W2[(size_t)(k + kb) * HID + colA];
        w0.y = W2[(size_t)(k + kb + 1) * HID + colA];
        w1.x = W2[(size_t)(k + kb) * HID + colB];
        w1.y = W2[(size_t)(k + kb + 1) * HID + colB];
        acc2[0][0] = wmma_f32(a0, w0, acc2[0][0]);
        acc2[1][0] = wmma_f32(a1, w0, acc2[1][0]);
        acc2[0][1] = wmma_f32(a0, w1, acc2[0][1]);
        acc2[1][1] = wmma_f32(a1, w1, acc2[1][1]);
    }

    // ---- Epilogue 2: + b2 + hh residual (hh still live in smem[:,0:256]) ----
#pragma unroll
    for (int rt = 0; rt < 2; ++rt) {
#pragma unroll
        for (int ct = 0; ct < 2; ++ct) {
            const int   col = (ct == 0) ? colA : colB;
            const float bb  = (ct == 0) ? b2a : b2b;
#pragma unroll
            for (int r = 0; r < 8; ++r) {
                int row = rt * 16 + r + 8 * hhalf;
                int g   = r0 + row;
                if (g < n) {
                    float v = acc2[rt][ct][r] + bb + smem[row][col];
                    __builtin_nontemporal_store(v, out_h + (size_t)g * HID + col);
                }
            }
        }
    }
}

extern "C" void kernel_launch(void* const* d_in, const int* in_sizes, int n_in,
                              void* d_out, int out_size, void* d_ws, size_t ws_size,
                              hipStream_t stream) {
    const float* x     = (const float*)d_in[0];
    const float* trans = (const float*)d_in[1];
    const float* ef    = (const float*)d_in[2];
    const float* hh    = (const float*)d_in[3];
    const float* W1    = (const float*)d_in[4];
    const float* b1    = (const float*)d_in[5];
    const float* W2    = (const float*)d_in[6];
    const float* b2    = (const float*)d_in[7];

    const int n = in_sizes[0] / 3;           // N from x:[N,3]
    float* coord = (float*)d_out;            // first N*3 floats
    float* out_h = (float*)d_out + (size_t)n * 3;

    coord_kernel<<<(n + 255) / 256, 256, 0, stream>>>(x, trans, coord, n);

    const int nblk = (n + ROWS_PER_BLK - 1) / ROWS_PER_BLK;
    mlp_kernel<<<nblk, 256, 0, stream>>>(ef, hh, W1, b1, W2, b2, out_h, n);
}